// Sampler_33363305955978
// MI455X (gfx1250) — compile-verified
//
#include <hip/hip_runtime.h>
#include <hip/hip_bf16.h>
#include <stdint.h>

#ifndef __has_builtin
#define __has_builtin(x) 0
#endif

#if __has_builtin(__builtin_amdgcn_global_load_async_to_lds_b32)
#define HAVE_ASYNC_B32 1
#else
#define HAVE_ASYNC_B32 0
#endif
#if __has_builtin(__builtin_amdgcn_s_wait_asynccnt)
#define HAVE_WAIT_ASYNC 1
#else
#define HAVE_WAIT_ASYNC 0
#endif

constexpr int S_ROWS = 512;
constexpr int VOCAB  = 50257;
constexpr int P_LEN  = 512;
constexpr int O_LEN  = 128;
constexpr int BLOCK  = 512;          // 16 waves (wave32)
constexpr int NWAVE  = BLOCK / 32;
constexpr int CNTW   = (VOCAB + 3) / 4;    // packed u8 counts, 4 per dword
constexpr int PBW    = (VOCAB + 31) / 32;  // prompt bitmask words
constexpr int HN     = 2048;               // radix histogram buckets
constexpr int TIE_CAP = 128;

static_assert(BLOCK == P_LEN, "one prompt token per thread");

struct Ctl {
  float M, Z, pRem;
  unsigned kRem;
  unsigned pfx;
  int bindPos;
  unsigned selBucket, selCA, selCnt;
  float selMA, selMass;
  unsigned cutKey;
  int nKeepTies, keepAllTies, tieN;
  float redF[NWAVE];
  unsigned redU[NWAVE];
};

union Smem {
  struct { unsigned cnt[CNTW]; unsigned pb[PBW]; } a;   // phase 1: penalty tables (~56.5KB)
  struct { unsigned hcnt[HN]; float hmass[HN]; } b;     // phase 2+: radix histograms (16KB)
};

// Monotone key: ascending uint order == ascending float order.
__device__ __forceinline__ unsigned fkey(float x) {
  unsigned u = __float_as_uint(x);
  return (u & 0x80000000u) ? ~u : (u | 0x80000000u);
}

__device__ __forceinline__ void block_max_to(float v, Ctl* c, int tid) {
  for (int o = 16; o > 0; o >>= 1) v = fmaxf(v, __shfl_xor(v, o, 32));
  if ((tid & 31) == 0) c->redF[tid >> 5] = v;
  __syncthreads();
  if (tid == 0) {
    float m = c->redF[0];
    for (int i = 1; i < NWAVE; ++i) m = fmaxf(m, c->redF[i]);
    c->M = m;
  }
  __syncthreads();
}

__device__ __forceinline__ void block_sum_to(float v, Ctl* c, int tid) {
  for (int o = 16; o > 0; o >>= 1) v += __shfl_xor(v, o, 32);
  if ((tid & 31) == 0) c->redF[tid >> 5] = v;
  __syncthreads();
  if (tid == 0) {
    float s = 0.f;
    for (int i = 0; i < NWAVE; ++i) s += c->redF[i];
    c->Z = s;
  }
  __syncthreads();
}

// Descending-order bucket select: finds the highest bucket where the top-k
// count budget or the top-p mass budget binds. Binding is suffix-monotone,
// so atomicMin over descending positions yields the first binding bucket.
__device__ void block_select(const unsigned* hcnt, const float* hmass, int nb,
                             unsigned kB, float pB, Ctl* c, int tid)
{
  const int CH = nb / BLOCK;          // buckets per thread (4 or 2)
  unsigned myc[4]; float mym[4];
  unsigned lc = 0u; float lm = 0.f;
  for (int q = 0; q < CH; ++q) {
    int b = nb - 1 - (tid * CH + q);  // descending positions
    myc[q] = hcnt[b]; mym[q] = hmass[b];
    lc += myc[q]; lm += mym[q];
  }
  const int lane = tid & 31;
  const int w = tid >> 5;
  unsigned ic = lc; float im = lm;    // wave inclusive scan
  for (int o = 1; o < 32; o <<= 1) {
    unsigned uc = __shfl_up(ic, o, 32);
    float um = __shfl_up(im, o, 32);
    if (lane >= o) { ic += uc; im += um; }
  }
  if (tid == 0) c->bindPos = 0x7FFFFFFF;
  if (lane == 31) { c->redU[w] = ic; c->redF[w] = im; }
  __syncthreads();
  if (tid < NWAVE) {                  // exclusive scan of wave totals
    unsigned v = c->redU[tid]; float f = c->redF[tid];
    unsigned ov = v; float of = f;
    for (int o = 1; o < NWAVE; o <<= 1) {
      unsigned uv = __shfl_up(v, o, 32);
      float uf = __shfl_up(f, o, 32);
      if (tid >= o) { v += uv; f += uf; }
    }
    c->redU[tid] = v - ov;
    c->redF[tid] = f - of;
  }
  __syncthreads();
  unsigned baseC = c->redU[w] + (ic - lc);  // exclusive "above" totals at first owned pos
  float    baseM = c->redF[w] + (im - lm);
  int firstPos = 0x7FFFFFFF;
  {
    unsigned rc = baseC; float rm = baseM;
    for (int q = 0; q < CH; ++q) {
      if ((rc + myc[q] >= kB) || (rm + mym[q] > pB)) { firstPos = tid * CH + q; break; }
      rc += myc[q]; rm += mym[q];
    }
  }
  if (firstPos != 0x7FFFFFFF) atomicMin(&c->bindPos, firstPos);
  __syncthreads();
  if (tid == 0 && c->bindPos == 0x7FFFFFFF) c->bindPos = nb - 1;  // safety (should not happen)
  __syncthreads();
  const int bp = c->bindPos;
  if (bp >= tid * CH && bp < tid * CH + CH) {
    unsigned rc = baseC; float rm = baseM;
    for (int q = 0; q < CH; ++q) {
      int p = tid * CH + q;
      if (p == bp) {
        c->selBucket = (unsigned)(nb - 1 - p);
        c->selCA = rc; c->selMA = rm;
        c->selCnt = myc[q]; c->selMass = mym[q];
        break;
      }
      rc += myc[q]; rm += mym[q];
    }
  }
  __syncthreads();
}

__global__ __launch_bounds__(BLOCK) void sampler_kernel(
    const float* __restrict__ logits,
    const int*   __restrict__ prompt,
    const int*   __restrict__ outtok,
    const float* __restrict__ presence,
    const float* __restrict__ frequency,
    const float* __restrict__ repetition,
    const float* __restrict__ top_ps,
    const int*   __restrict__ top_ks,
    const float* __restrict__ min_ps,
    float*       __restrict__ out)
{
  __shared__ Smem sm;
  __shared__ Ctl ctl;
  __shared__ unsigned tieList[TIE_CAP];
  __shared__ int tokBuf[P_LEN + O_LEN];

  const int tid = threadIdx.x;
  const int row = blockIdx.x;
  const float* lr = logits + (size_t)row * VOCAB;
  float* orow = out + (size_t)row * VOCAB;

  // ---- token fetch: async bulk copy to LDS (CDNA5 ASYNCcnt path) ----
  const int* pr = prompt + (size_t)row * P_LEN;
  const int* ot = outtok + (size_t)row * O_LEN;
#if HAVE_ASYNC_B32
  __builtin_amdgcn_global_load_async_to_lds_b32((int*)(pr + tid), (int*)&tokBuf[tid], 0, 0);
  if (tid < O_LEN) {
    __builtin_amdgcn_global_load_async_to_lds_b32((int*)(ot + tid), (int*)&tokBuf[P_LEN + tid], 0, 0);
  }
#else
  tokBuf[tid] = pr[tid];
  if (tid < O_LEN) tokBuf[P_LEN + tid] = ot[tid];
#endif

  // Zero penalty tables while the async copies fly.
  for (int i = tid; i < CNTW; i += BLOCK) sm.a.cnt[i] = 0u;
  for (int i = tid; i < PBW;  i += BLOCK) sm.a.pb[i]  = 0u;

#if HAVE_ASYNC_B32
# if HAVE_WAIT_ASYNC
  __builtin_amdgcn_s_wait_asynccnt(0);
# else
  asm volatile("s_wait_asynccnt 0" ::: "memory");
# endif
#endif
  __syncthreads();

  // Scatter: prompt -> bitmask; output -> packed u8 bincount (max 128 fits a byte).
  {
    int t1 = tokBuf[tid];
    atomicOr(&sm.a.pb[t1 >> 5], 1u << (t1 & 31));
    if (tid < O_LEN) {
      int t2 = tokBuf[P_LEN + tid];
      atomicAdd(&sm.a.cnt[t2 >> 2], 1u << (8 * (t2 & 3)));
    }
  }
  __syncthreads();

  const float repv  = repetition[row];
  const float freqv = frequency[row];
  const float presv = presence[row];
  const float toppv = top_ps[row];
  const int   topkv = top_ks[row];
  const float minpv = min_ps[row];
  const float NEG_INF = -__builtin_huge_valf();

  // ---- Pass 1: penalties; d_out doubles as the penalized-logits scratch ----
  float vmax = NEG_INF;
  for (int i = tid; i < VOCAB; i += BLOCK) {
    __builtin_prefetch(lr + i + 4 * BLOCK, 0, 0);
    float L = lr[i];
    unsigned c8 = (sm.a.cnt[i >> 2] >> (8 * (i & 3))) & 0xFFu;
    bool pm = (sm.a.pb[i >> 5] >> (i & 31)) & 1u;
    bool seen = pm || (c8 != 0u);
    float x = L;
    if (seen) x = (L > 0.f) ? (L / repv) : (L * repv);
    x -= freqv * (float)c8;
    if (c8 != 0u) x -= presv;
    orow[i] = x;
    vmax = fmaxf(vmax, x);
  }
  block_max_to(vmax, &ctl, tid);
  const float Mv = ctl.M;
  __syncthreads();   // sm.a dead from here; sm.b (histogram) takes over

  // ---- Level 0: histogram of key[31:21]; Z falls out of the mass totals ----
  for (int i = tid; i < HN; i += BLOCK) { sm.b.hcnt[i] = 0u; sm.b.hmass[i] = 0.f; }
  __syncthreads();
  for (int i = tid; i < VOCAB; i += BLOCK) {
    float x = orow[i];
    unsigned k = fkey(x);
    atomicAdd(&sm.b.hcnt[k >> 21], 1u);
    atomicAdd(&sm.b.hmass[k >> 21], __expf(x - Mv));
  }
  __syncthreads();
  {
    float zp = 0.f;
    for (int i = tid; i < HN; i += BLOCK) zp += sm.b.hmass[i];
    block_sum_to(zp, &ctl, tid);
    if (tid == 0) { ctl.pRem = toppv * ctl.Z; ctl.kRem = (unsigned)topkv; }
    __syncthreads();
    unsigned kB = ctl.kRem; float pB = ctl.pRem;
    block_select(sm.b.hcnt, sm.b.hmass, HN, kB, pB, &ctl, tid);
    if (tid == 0) {
      ctl.pfx  = ctl.selBucket << 21;
      ctl.kRem -= ctl.selCA;
      ctl.pRem -= ctl.selMA;
    }
    __syncthreads();
  }

  // ---- Level 1: key[20:10] within the chosen L0 bucket ----
  for (int i = tid; i < HN; i += BLOCK) { sm.b.hcnt[i] = 0u; sm.b.hmass[i] = 0.f; }
  __syncthreads();
  {
    unsigned pfx = ctl.pfx;
    for (int i = tid; i < VOCAB; i += BLOCK) {
      float x = orow[i];
      unsigned k = fkey(x);
      if ((k >> 21) == (pfx >> 21)) {
        unsigned b = (k >> 10) & 2047u;
        atomicAdd(&sm.b.hcnt[b], 1u);
        atomicAdd(&sm.b.hmass[b], __expf(x - Mv));
      }
    }
    __syncthreads();
    unsigned kB = ctl.kRem; float pB = ctl.pRem;
    block_select(sm.b.hcnt, sm.b.hmass, HN, kB, pB, &ctl, tid);
    if (tid == 0) {
      ctl.pfx  |= ctl.selBucket << 10;
      ctl.kRem -= ctl.selCA;
      ctl.pRem -= ctl.selMA;
    }
    __syncthreads();
  }

  // ---- Level 2: key[9:0] -> exact 32-bit cutoff key ----
  for (int i = tid; i < 1024; i += BLOCK) { sm.b.hcnt[i] = 0u; sm.b.hmass[i] = 0.f; }
  __syncthreads();
  {
    unsigned pfx = ctl.pfx;
    for (int i = tid; i < VOCAB; i += BLOCK) {
      float x = orow[i];
      unsigned k = fkey(x);
      if ((k >> 10) == (pfx >> 10)) {
        unsigned b = k & 1023u;
        atomicAdd(&sm.b.hcnt[b], 1u);
        atomicAdd(&sm.b.hmass[b], __expf(x - Mv));
      }
    }
    __syncthreads();
    unsigned kB = ctl.kRem; float pB = ctl.pRem;
    block_select(sm.b.hcnt, sm.b.hmass, 1024, kB, pB, &ctl, tid);
  }

  // ---- Tie budgets at the exact cutoff value ----
  if (tid == 0) {
    ctl.kRem -= ctl.selCA;
    ctl.pRem -= ctl.selMA;
    ctl.cutKey = ctl.pfx | ctl.selBucket;
    unsigned cntT = ctl.selCnt;
    float eT = (cntT > 0u) ? (ctl.selMass / (float)cntT) : 0.f;
    int nK = (int)ctl.kRem;                   // >= 1 by construction
    int nP;
    float R = ctl.pRem;                       // >= 0 by construction
    if (!(eT > 0.f)) {
      nP = (int)cntT;                         // zero-mass ties always pass top-p
    } else {
      float r = R / eT;                       // j kept iff j*eT <= R
      nP = (r >= (float)cntT) ? (int)cntT : ((int)r + 1);
    }
    int nKeep = nK < nP ? nK : nP;
    if (nKeep < 1) nKeep = 1;
    if (nKeep > (int)cntT) nKeep = (int)cntT;
    ctl.nKeepTies = nKeep;
    ctl.keepAllTies = (nKeep >= (int)cntT) ? 1 : 0;
    ctl.tieN = 0;
  }
  __syncthreads();

  // ---- Rare path: index-stable selection among exact-value ties ----
  if (!ctl.keepAllTies) {
    unsigned ck = ctl.cutKey;
    for (int i = tid; i < VOCAB; i += BLOCK) {
      float x = orow[i];
      if (fkey(x) == ck) {
        int p = atomicAdd(&ctl.tieN, 1);
        if (p < TIE_CAP) tieList[p] = (unsigned)i;
      }
    }
    __syncthreads();
    if (tid == 0) {
      if (ctl.tieN > TIE_CAP) {
        ctl.keepAllTies = 1;                  // pathological overflow: keep all ties
      } else {
        int n = ctl.tieN;                     // insertion sort ascending (tiny n)
        for (int a2 = 1; a2 < n; ++a2) {
          unsigned key = tieList[a2];
          int b2 = a2 - 1;
          while (b2 >= 0 && tieList[b2] > key) { tieList[b2 + 1] = tieList[b2]; --b2; }
          tieList[b2 + 1] = key;
        }
      }
    }
    __syncthreads();
  }

  // ---- Final pass: top-p/top-k cutoff + algebraic min-p (exp(x-M) < min_p) ----
  {
    const unsigned ck = ctl.cutKey;
    const int keepAll = ctl.keepAllTies;
    const int nKeep = ctl.nKeepTies;
    for (int i = tid; i < VOCAB; i += BLOCK) {
      float x = orow[i];
      unsigned k = fkey(x);
      bool keep;
      if (k > ck) {
        keep = true;
      } else if (k == ck) {
        if (keepAll) {
          keep = true;
        } else {
          keep = false;
          for (int j = 0; j < nKeep; ++j)
            if (tieList[j] == (unsigned)i) { keep = true; break; }
        }
      } else {
        keep = false;
      }
      if (keep && (__expf(x - Mv) < minpv)) keep = false;
      orow[i] = keep ? x : NEG_INF;
    }
  }
}

extern "C" void kernel_launch(void* const* d_in, const int* in_sizes, int n_in,
                              void* d_out, int out_size, void* d_ws, size_t ws_size,
                              hipStream_t stream) {
  (void)n_in; (void)out_size; (void)d_ws; (void)ws_size;
  const float* logits  = (const float*)d_in[0];
  const int*   prompt  = (const int*)d_in[1];
  const int*   outtok  = (const int*)d_in[2];
  const float* presen  = (const float*)d_in[3];
  const float* freq    = (const float*)d_in[4];
  const float* rep     = (const float*)d_in[5];
  const float* topp    = (const float*)d_in[6];
  const int*   topk    = (const int*)d_in[7];
  const float* minp    = (const float*)d_in[8];
  float* out = (float*)d_out;

  int rows = in_sizes[3] > 0 ? in_sizes[3] : S_ROWS;   // one block per row
  sampler_kernel<<<rows, BLOCK, 0, stream>>>(
      logits, prompt, outtok, presen, freq, rep, topp, topk, minp, out);
}